// SSIMLoss_69561290326635
// MI455X (gfx1250) — compile-verified
//
#include <hip/hip_runtime.h>
#include <math.h>

typedef __attribute__((ext_vector_type(2))) float v2f;
typedef __attribute__((ext_vector_type(8))) float v8f;

#define TSR   32      // output tile rows per workgroup
#define TSC   64      // output tile cols per workgroup
#define IR    48      // padded input-tile rows (3 x 16 WMMA row-bands)
#define IC    76      // padded input-tile cols; col ci <-> global col (c0 - 6 + ci)
#define SS    65      // intermediate S stride (odd -> fewer bank conflicts)
#define IMH   512
#define IMW   512
#define NTHR  256     // 8 wave32

// select w[idx] for idx in [0,10], else 0 — unrolled cndmask chain, stays in VGPRs
__device__ __forceinline__ float bandsel(int idx, const float* w) {
    float v = 0.f;
#pragma unroll
    for (int i = 0; i < 11; ++i) v = (idx == i) ? w[i] : v;
    return v;
}

__global__ __launch_bounds__(NTHR) void ssim_stage1(
    const float* __restrict__ pred, const float* __restrict__ targ,
    const float* __restrict__ window, float* __restrict__ partials)
{
    __shared__ __align__(16) float ldsP[IR * IC];        // 14.6 KB
    __shared__ __align__(16) float ldsT[IR * IC];        // 14.6 KB
    __shared__ __align__(16) float ldsS[5 * IR * SS];    // 62.4 KB (5 h-blurred maps)
    __shared__ __align__(16) float ldsWH[7 * 64];        // horiz frags: w1[k-n-1]
    __shared__ __align__(16) float ldsWV[7 * 64];        // vert  frags: w1[k-m]
    __shared__ __align__(16) float ldsRed[NTHR];

    const int tid  = threadIdx.x;
    const int lane = tid & 31;
    // scalarize the wave id so task decode / map select stay on the SALU
    const int wave = __builtin_amdgcn_readfirstlane(tid >> 5);
    const int n16  = lane & 15;
    const int hi2  = (lane >> 4) << 1;        // 0 for lanes 0-15, 2 for lanes 16-31

    const int r0 = blockIdx.y * TSR;
    const int c0 = blockIdx.x * TSC;
    const size_t imgBase = (size_t)blockIdx.z * ((size_t)IMH * IMW);

    // ---- load zero-padded halo tiles as aligned float2 (gc = c0-6+2j is even,
    //      and 512 is even, so both elements are in/out of bounds together) ----
    for (int i = tid; i < IR * (IC / 2); i += NTHR) {
        int r = i / (IC / 2), j = i % (IC / 2);
        int gr = r0 - 5 + r;
        int gc = c0 - 6 + 2 * j;
        float2 vp = make_float2(0.f, 0.f);
        float2 vt = make_float2(0.f, 0.f);
        if ((gr >= 0) && (gr < IMH) && (gc >= 0) && (gc < IMW)) {
            size_t g = imgBase + (size_t)gr * IMW + gc;
            vp = *(const float2*)&pred[g];
            vt = *(const float2*)&targ[g];
        }
        *(float2*)&ldsP[r * IC + 2 * j] = vp;
        *(float2*)&ldsT[r * IC + 2 * j] = vt;
    }

    // ---- recover 1-D gaussian from window center row: w1[i] = W[5][i] / sqrt(W[5][5]) ----
    float w1[11];
    {
        float w5 = sqrtf(window[60]);
        #pragma unroll
        for (int i = 0; i < 11; ++i) w1[i] = window[55 + i] / w5;
    }
    // ---- banded weight fragments (A and B lane layouts are mutual transposes):
    //      horizontal B:  WbH[k][n] = w1[k-n-1]   (data window starts at c0-6)
    //      vertical   A:  WaV[m][k] = w1[k-m]     (S rows start at r0-5)      ----
    if (wave == 0) {
        #pragma unroll
        for (int q = 0; q < 7; ++q) {
            int k0 = 4 * q + hi2;
            ldsWH[q * 64 + lane * 2 + 0] = bandsel(k0     - n16 - 1, w1);
            ldsWH[q * 64 + lane * 2 + 1] = bandsel(k0 + 1 - n16 - 1, w1);
            ldsWV[q * 64 + lane * 2 + 0] = bandsel(k0     - n16,     w1);
            ldsWV[q * 64 + lane * 2 + 1] = bandsel(k0 + 1 - n16,     w1);
        }
    }
    __syncthreads();

    // ============ horizontal pass: Out(16x16) = In(16x28) x WbH(28x16) ============
    // tasks: 5 maps x 3 row-bands x 4 col-tiles = 60, round-robin over 8 waves
    for (int task = wave; task < 60; task += 8) {
        const int mi   = task / 12;          // scalar
        const int rem  = task % 12;
        const int band = rem >> 2;
        const int ct   = rem & 3;

        v8f acc = {};
        const int abase = (band * 16 + n16) * IC + ct * 16 + hi2;   // even -> b64 loads
        #pragma unroll
        for (int q = 0; q < 7; ++q) {
            v2f a;
            if (mi == 0) {
                const float2 ap = *(const float2*)&ldsP[abase + 4 * q];
                a.x = ap.x; a.y = ap.y;
            } else if (mi == 1) {
                const float2 at = *(const float2*)&ldsT[abase + 4 * q];
                a.x = at.x; a.y = at.y;
            } else if (mi == 2) {
                const float2 ap = *(const float2*)&ldsP[abase + 4 * q];
                a.x = ap.x * ap.x; a.y = ap.y * ap.y;
            } else if (mi == 3) {
                const float2 at = *(const float2*)&ldsT[abase + 4 * q];
                a.x = at.x * at.x; a.y = at.y * at.y;
            } else {
                const float2 ap = *(const float2*)&ldsP[abase + 4 * q];
                const float2 at = *(const float2*)&ldsT[abase + 4 * q];
                a.x = ap.x * at.x; a.y = ap.y * at.y;
            }
            const v2f b = *(const v2f*)&ldsWH[q * 64 + lane * 2];
            acc = __builtin_amdgcn_wmma_f32_16x16x4_f32(
                false, a, false, b, (short)0, acc, false, false);
        }
        // D layout: VGPR v -> row M=v (lanes 0-15) / M=v+8 (lanes 16-31)
        const int sr = band * 16 + ((lane >= 16) ? 8 : 0);
        const int sc = ct * 16 + n16;
        float* S = &ldsS[mi * (IR * SS)];
        #pragma unroll
        for (int v = 0; v < 8; ++v) S[(sr + v) * SS + sc] = acc[v];
    }
    __syncthreads();

    // ============ vertical pass: Out(16x16) = WaV(16x28) x S(28x16) + fused SSIM ============
    const int qr  = wave >> 2;               // scalar: 2 row-quadrants
    const int qc  = wave & 3;                // scalar: 4 col-quadrants -> 8 waves exactly
    const int col = qc * 16 + n16;

    v8f m1 = {}, m2 = {}, bpp = {}, btt = {}, bpt = {};
    #pragma unroll
    for (int q = 0; q < 7; ++q) {
        const v2f aw = *(const v2f*)&ldsWV[q * 64 + lane * 2];
        const int s0 = qr * 16 + 4 * q + hi2;
        const int o0 = s0 * SS + col;
        v2f b;
        b.x = ldsS[0 * (IR * SS) + o0]; b.y = ldsS[0 * (IR * SS) + o0 + SS];
        m1  = __builtin_amdgcn_wmma_f32_16x16x4_f32(false, aw, false, b, (short)0, m1,  false, false);
        b.x = ldsS[1 * (IR * SS) + o0]; b.y = ldsS[1 * (IR * SS) + o0 + SS];
        m2  = __builtin_amdgcn_wmma_f32_16x16x4_f32(false, aw, false, b, (short)0, m2,  false, false);
        b.x = ldsS[2 * (IR * SS) + o0]; b.y = ldsS[2 * (IR * SS) + o0 + SS];
        bpp = __builtin_amdgcn_wmma_f32_16x16x4_f32(false, aw, false, b, (short)0, bpp, false, false);
        b.x = ldsS[3 * (IR * SS) + o0]; b.y = ldsS[3 * (IR * SS) + o0 + SS];
        btt = __builtin_amdgcn_wmma_f32_16x16x4_f32(false, aw, false, b, (short)0, btt, false, false);
        b.x = ldsS[4 * (IR * SS) + o0]; b.y = ldsS[4 * (IR * SS) + o0 + SS];
        bpt = __builtin_amdgcn_wmma_f32_16x16x4_f32(false, aw, false, b, (short)0, bpt, false, false);
    }

    // ---- pointwise SSIM on the 8 accumulator elements of this lane ----
    const float C1 = 1e-4f, C2 = 9e-4f, EPS = 1e-8f;
    float lsum = 0.f;
    #pragma unroll
    for (int v = 0; v < 8; ++v) {
        float mu1 = m1[v], mu2 = m2[v];
        float mu1sq = mu1 * mu1, mu2sq = mu2 * mu2, mu12 = mu1 * mu2;
        float s1  = bpp[v] - mu1sq;
        float s2  = btt[v] - mu2sq;
        float s12 = bpt[v] - mu12;
        float num = (2.f * mu12 + C1) * (2.f * s12 + C2);
        float den = (mu1sq + mu2sq + C1) * (s1 + s2 + C2) + EPS;
        lsum += num / den;
    }

    // ---- deterministic in-block reduction ----
    ldsRed[tid] = lsum;
    __syncthreads();
    #pragma unroll
    for (int off = NTHR / 2; off > 0; off >>= 1) {
        if (tid < off) ldsRed[tid] += ldsRed[tid + off];
        __syncthreads();
    }
    if (tid == 0) {
        size_t bid = ((size_t)blockIdx.z * gridDim.y + blockIdx.y) * gridDim.x + blockIdx.x;
        partials[bid] = ldsRed[0];
    }
}

__global__ __launch_bounds__(256) void ssim_stage2(
    const float* __restrict__ partials, float* __restrict__ out,
    int n, double invN)
{
    __shared__ double sd[256];
    double s = 0.0;
    for (int i = threadIdx.x; i < n; i += 256) s += (double)partials[i];
    sd[threadIdx.x] = s;
    __syncthreads();
    for (int off = 128; off > 0; off >>= 1) {
        if (threadIdx.x < off) sd[threadIdx.x] += sd[threadIdx.x + off];
        __syncthreads();
    }
    if (threadIdx.x == 0) out[0] = (float)(1.0 - sd[0] * invN);
}

extern "C" void kernel_launch(void* const* d_in, const int* in_sizes, int n_in,
                              void* d_out, int out_size, void* d_ws, size_t ws_size,
                              hipStream_t stream) {
    const float* pred   = (const float*)d_in[0];
    const float* targ   = (const float*)d_in[1];
    const float* window = (const float*)d_in[2];
    float* out      = (float*)d_out;
    float* partials = (float*)d_ws;

    const int BC = in_sizes[0] / (IMH * IMW);   // 48 channel-maps folded into batch
    dim3 grid(IMW / TSC, IMH / TSR, BC);        // 8 x 16 x 48 blocks
    ssim_stage1<<<grid, NTHR, 0, stream>>>(pred, targ, window, partials);

    const int nPart = (IMW / TSC) * (IMH / TSR) * BC;
    const double invN = 1.0 / ((double)BC * (double)IMH * (double)IMW);
    ssim_stage2<<<1, 256, 0, stream>>>(partials, out, nPart, invN);
}